// BernConv_72645076845144
// MI455X (gfx1250) — compile-verified
//
#include <hip/hip_runtime.h>

typedef __attribute__((ext_vector_type(2))) float v2f;
typedef __attribute__((ext_vector_type(8))) float v8f;

#define D 256            // D_IN == D_OUT == 256

// ---------------------------------------------------------------------------
// Repack weight [256][256] (row major, K x N) into WMMA B-fragment order.
// For k-step ks (K = 4*ks..4*ks+3) and column tile ct (N = 16*ct..16*ct+15):
//   lane < 16 : VGPR0 = B[4ks+0][ct*16+lane], VGPR1 = B[4ks+1][ct*16+lane]
//   lane >= 16: VGPR0 = B[4ks+2][ct*16+lane-16], VGPR1 = B[4ks+3][ct*16+lane-16]
// Dest layout: wpack[ ((ks*16 + ct)*32 + lane)*2 + j ]
// ---------------------------------------------------------------------------
__global__ void repack_weight_kernel(const float* __restrict__ w,
                                     float* __restrict__ wpack) {
    int t = blockIdx.x * blockDim.x + threadIdx.x;   // 0 .. 65535
    if (t >= D * D) return;
    int j    = t & 1;
    int lane = (t >> 1) & 31;
    int ct   = (t >> 6) & 15;
    int ks   = t >> 10;
    int krow = 4 * ks + 2 * (lane >> 4) + j;
    int col  = ct * 16 + (lane & 15);
    wpack[t] = w[krow * D + col];
}

// ---------------------------------------------------------------------------
// support = x @ weight via V_WMMA_F32_16X16X4_F32.
// Block = 256 threads = 8 waves. Each block: 32 rows x 256 cols.
//   wave&3  -> column-tile group (cols 64*(wave&3) .. +63)
//   wave>>2 -> row half (rows row0 .. +15 or row0+16 .. +31)
// A panel (32 x 256 fp32 = 32 KB) is DMA'd into LDS with
// GLOBAL_LOAD_ASYNC_TO_LDS_B128 (ASYNCcnt), then read with ds_load_b64.
// Writes the result to BOTH ws_support (hop source) and d_out (accumulator).
// ---------------------------------------------------------------------------
__global__ void __launch_bounds__(256)
gemm_wmma_kernel(const float* __restrict__ x,
                 const float* __restrict__ wpack,
                 float* __restrict__ support,
                 float* __restrict__ out) {
    __shared__ float As[32 * D];   // 32 KB

    const int tid  = threadIdx.x;
    const int row0 = blockIdx.x * 32;

    // Async DMA of A panel: 32 rows * 64 float4 = 2048 float4; 8 per thread.
    {
        const float4* xsrc = (const float4*)(x + (size_t)row0 * D);
        const unsigned ldsBase = (unsigned)(uintptr_t)(&As[0]);
#pragma unroll
        for (int i = 0; i < 8; ++i) {
            int idx = i * 256 + tid;
            unsigned     ldsoff = ldsBase + idx * 16;
            const float4* gaddr = xsrc + idx;
            asm volatile("global_load_async_to_lds_b128 %0, %1, off"
                         :: "v"(ldsoff), "v"(gaddr) : "memory");
        }
        asm volatile("s_wait_asynccnt 0x0" ::: "memory");
    }
    __syncthreads();

    const int wave    = tid >> 5;        // 0..7
    const int lane    = tid & 31;
    const int lane15  = lane & 15;
    const int laneHi  = lane >> 4;       // 0 or 1
    const int rowHalf = wave >> 2;       // 0 or 1
    const int bct     = (wave & 3) * 4;  // first column tile for this wave

    v8f acc0 = {}, acc1 = {}, acc2 = {}, acc3 = {};

    // Per-lane A fragment base offset in LDS (floats).
    const int aoff = (rowHalf * 16 + lane15) * D + 2 * laneHi;
    const v2f* __restrict__ bp = (const v2f*)wpack;

#pragma unroll 4
    for (int ks = 0; ks < D / 4; ++ks) {
        v2f a  = *(const v2f*)&As[aoff + 4 * ks];
        v2f b0 = bp[((ks * 16) + bct + 0) * 32 + lane];
        v2f b1 = bp[((ks * 16) + bct + 1) * 32 + lane];
        v2f b2 = bp[((ks * 16) + bct + 2) * 32 + lane];
        v2f b3 = bp[((ks * 16) + bct + 3) * 32 + lane];
        acc0 = __builtin_amdgcn_wmma_f32_16x16x4_f32(false, a, false, b0,
                                                     (short)0, acc0, false, false);
        acc1 = __builtin_amdgcn_wmma_f32_16x16x4_f32(false, a, false, b1,
                                                     (short)0, acc1, false, false);
        acc2 = __builtin_amdgcn_wmma_f32_16x16x4_f32(false, a, false, b2,
                                                     (short)0, acc2, false, false);
        acc3 = __builtin_amdgcn_wmma_f32_16x16x4_f32(false, a, false, b3,
                                                     (short)0, acc3, false, false);
    }

    // Store: C/D layout — VGPR r, lane<16 -> (M=r, N=lane); lane>=16 -> (M=8+r, N=lane-16).
    v8f accs[4] = {acc0, acc1, acc2, acc3};
#pragma unroll
    for (int ctl = 0; ctl < 4; ++ctl) {
        int col = (bct + ctl) * 16 + lane15;
#pragma unroll
        for (int r = 0; r < 8; ++r) {
            int row = row0 + rowHalf * 16 + r + 8 * laneHi;
            size_t idx = (size_t)row * D + col;
            float v = accs[ctl][r];
            support[idx] = v;
            out[idx]     = v;
        }
    }
}

// ---------------------------------------------------------------------------
// SpMM hop: dst[row_e] += w_e * src[col_e]   (atomic scatter, L2-resident)
// 64 lanes per edge, 4 edges per 256-thread block, float4 per lane.
// ---------------------------------------------------------------------------
__global__ void __launch_bounds__(256)
spmm_kernel(const float* __restrict__ ew,
            const int* __restrict__ er,
            const int* __restrict__ ec,
            const float* __restrict__ src,
            float* __restrict__ dst,
            int E) {
    int e = blockIdx.x * 4 + (threadIdx.x >> 6);
    if (e >= E) return;
    int lane = threadIdx.x & 63;

    float w = ew[e];
    int   r = er[e];
    int   c = ec[e];

    const float4 v = ((const float4*)(src + (size_t)c * D))[lane];
    float* o = dst + (size_t)r * D + lane * 4;
    atomicAdd(o + 0, w * v.x);
    atomicAdd(o + 1, w * v.y);
    atomicAdd(o + 2, w * v.z);
    atomicAdd(o + 3, w * v.w);
}

// ---------------------------------------------------------------------------
// Elementwise helpers (float4 granularity).
// ---------------------------------------------------------------------------
__global__ void zero_kernel(float4* __restrict__ z, int n4) {
    int i = blockIdx.x * blockDim.x + threadIdx.x;
    if (i < n4) z[i] = make_float4(0.f, 0.f, 0.f, 0.f);
}

// out += add; z = 0   (fused accumulate + prepare next hop target)
__global__ void acc_zero_kernel(float4* __restrict__ out,
                                const float4* __restrict__ add,
                                float4* __restrict__ z, int n4) {
    int i = blockIdx.x * blockDim.x + threadIdx.x;
    if (i >= n4) return;
    float4 o = out[i], a = add[i];
    o.x += a.x; o.y += a.y; o.z += a.z; o.w += a.w;
    out[i] = o;
    z[i] = make_float4(0.f, 0.f, 0.f, 0.f);
}

// out += add + bias (bias broadcast per row; 64 float4 per row)
__global__ void final_bias_kernel(float4* __restrict__ out,
                                  const float4* __restrict__ add,
                                  const float4* __restrict__ bias4, int n4) {
    int i = blockIdx.x * blockDim.x + threadIdx.x;
    if (i >= n4) return;
    float4 o = out[i], a = add[i], b = bias4[i & 63];
    o.x += a.x + b.x; o.y += a.y + b.y; o.z += a.z + b.z; o.w += a.w + b.w;
    out[i] = o;
}

// ---------------------------------------------------------------------------
// Launch sequence (deterministic, all on `stream`):
//   repack -> zero(B) -> gemm(support->A, out) ->
//   spmm(A->B) -> out+=B, zero(A) -> spmm(B->A) -> out+=A, zero(B) ->
//   spmm(A->B) -> out += B + bias
// ---------------------------------------------------------------------------
extern "C" void kernel_launch(void* const* d_in, const int* in_sizes, int n_in,
                              void* d_out, int out_size, void* d_ws, size_t ws_size,
                              hipStream_t stream) {
    const float* x      = (const float*)d_in[0];
    const float* weight = (const float*)d_in[1];
    const float* bias   = (const float*)d_in[2];
    const float* ew     = (const float*)d_in[3];
    const int*   er     = (const int*)d_in[4];
    const int*   ec     = (const int*)d_in[5];
    float*       out    = (float*)d_out;

    const int N  = in_sizes[0] / D;      // 100000
    const int E  = in_sizes[3];          // 3200000
    const int NT = N * D;                // total elements
    const int n4 = NT / 4;               // float4 count

    float* ws     = (float*)d_ws;
    float* bufA   = ws;                  // support / hop buffer A
    float* bufB   = ws + (size_t)NT;     // hop buffer B
    float* wpack  = bufB + (size_t)NT;   // repacked weight (65536 floats)

    const int ewBlocks = (n4 + 255) / 256;

    // 1. repack weight into B-fragment order
    repack_weight_kernel<<<(D * D + 255) / 256, 256, 0, stream>>>(weight, wpack);

    // 2. zero hop target B
    zero_kernel<<<ewBlocks, 256, 0, stream>>>((float4*)bufB, n4);

    // 3. support = x @ W  (written to bufA and out)
    gemm_wmma_kernel<<<N / 32, 256, 0, stream>>>(x, wpack, bufA, out);

    // 4. hop 1: B = spmm(A); out += B; zero A
    spmm_kernel<<<(E + 3) / 4, 256, 0, stream>>>(ew, er, ec, bufA, bufB, E);
    acc_zero_kernel<<<ewBlocks, 256, 0, stream>>>((float4*)out, (const float4*)bufB,
                                                  (float4*)bufA, n4);

    // 5. hop 2: A = spmm(B); out += A; zero B
    spmm_kernel<<<(E + 3) / 4, 256, 0, stream>>>(ew, er, ec, bufB, bufA, E);
    acc_zero_kernel<<<ewBlocks, 256, 0, stream>>>((float4*)out, (const float4*)bufA,
                                                  (float4*)bufB, n4);

    // 6. hop 3: B = spmm(A); out += B + bias
    spmm_kernel<<<(E + 3) / 4, 256, 0, stream>>>(ew, er, ec, bufA, bufB, E);
    final_bias_kernel<<<ewBlocks, 256, 0, stream>>>((float4*)out, (const float4*)bufB,
                                                    (const float4*)bias, n4);
}